// Network_57329223467783
// MI455X (gfx1250) — compile-verified
//
#include <hip/hip_runtime.h>
#include <math.h>

typedef float v2f __attribute__((ext_vector_type(2)));
typedef float v8f __attribute__((ext_vector_type(8)));

#define REEIG_EPS 1e-4f

enum { M_SQRT = 0, M_LOG = 1, M_EXP = 2, M_REEIG = 3 };

// ---------------------------------------------------------------------------
// fp32 WMMA 16x16x4: D = A(16x4) * B(4x16) + C.
// A layout (ISA 7.12.2, 32-bit A 16x4): lanes 0-15 -> M=lane, VGPR0=K0,VGPR1=K1;
// lanes 16-31 -> K2,K3.  B/C/D: row striped across lanes within a VGPR.
// ---------------------------------------------------------------------------
__device__ inline v8f wmma4(v2f a, v2f b, v8f c) {
  return __builtin_amdgcn_wmma_f32_16x16x4_f32(false, a, false, b, (short)0, c,
                                               false, false);
}

// ---------------------------------------------------------------------------
// Generic LDS-resident tiled matmul on WMMA f32 16x16x4.
// C[MT*16 x NT*16] = alpha * opA(A) * opB(B) + beta_diag * I
// opA(m,k) = TA ? A[k*lda+m] : A[m*lda+k]  (same for B).
// K unrolled by 16: 8 fragment loads hoisted ahead of 4 back-to-back WMMAs so
// the LDS round-trip is paid once per 4 matrix ops, not once per op.
// Tiles distributed over NW waves; tile predicate is wave-uniform (EXEC all-1).
// ---------------------------------------------------------------------------
template <int MT, int NT, int K, int NW, bool TA, bool TB>
__device__ void mmt(const float* A, int lda, const float* B, int ldb, float* C,
                    int ldc, float alpha, float beta_diag) {
  static_assert((K & 15) == 0, "K must be a multiple of 16");
  const int lane = threadIdx.x & 31;
  const int wave = threadIdx.x >> 5;
  const int half = lane >> 4;
  const int li = lane & 15;
  __syncthreads();
  for (int t = wave; t < MT * NT; t += NW) {
    const int ti = t / NT, tj = t % NT;
    const int m = ti * 16 + li;
    const int n = tj * 16 + li;
    v8f acc = {0.f, 0.f, 0.f, 0.f, 0.f, 0.f, 0.f, 0.f};
    for (int k = 0; k < K; k += 16) {
      v2f a[4], b[4];
#pragma unroll
      for (int u = 0; u < 4; ++u) {
        const int ku = k + 4 * u;
        const int ka = ku + 2 * half;
        a[u].x = TA ? A[(ka)*lda + m] : A[m * lda + ka];
        a[u].y = TA ? A[(ka + 1) * lda + m] : A[m * lda + ka + 1];
        const int kb0 = ku + half, kb1 = ku + 2 + half;
        b[u].x = TB ? B[n * ldb + kb0] : B[kb0 * ldb + n];
        b[u].y = TB ? B[n * ldb + kb1] : B[kb1 * ldb + n];
      }
#pragma unroll
      for (int u = 0; u < 4; ++u) acc = wmma4(a[u], b[u], acc);
    }
#pragma unroll
    for (int r = 0; r < 8; ++r) {
      const int row = ti * 16 + r + 8 * half;
      float v = alpha * acc[r];
      if (row == n) v += beta_diag;
      C[row * ldc + n] = v;
    }
  }
  __syncthreads();
}

// ------------------------- small LDS helpers -------------------------------
__device__ inline float block_sum(float part, float* red) {
  if (threadIdx.x == 0) *red = 0.f;
  __syncthreads();
  atomicAdd(red, part);
  __syncthreads();
  float r = *red;
  __syncthreads();
  return r;
}

template <int NDIAG, int NP>
__device__ inline float trace_n(const float* A, float* red) {
  float s = 0.f;
  for (int i = threadIdx.x; i < NDIAG; i += blockDim.x) s += A[i * NP + i];
  return block_sum(s, red);
}

template <int N, int NP>
__device__ inline void load_pad(float* dst, const float* src, float diag_pad) {
  for (int i = threadIdx.x; i < NP * NP; i += blockDim.x) {
    int r = i / NP, c = i % NP;
    dst[i] = (r < N && c < N) ? src[r * N + c] : ((r == c) ? diag_pad : 0.f);
  }
  __syncthreads();
}

template <int N, int NP>
__device__ inline void store_sym(float* __restrict__ dst, const float* S,
                                 float scale, float diag_add) {
  for (int i = threadIdx.x; i < N * N; i += blockDim.x) {
    int r = i / N, c = i % N;
    float v = 0.5f * (S[r * NP + c] + S[c * NP + r]);
    v = v * scale + ((r == c) ? diag_add : 0.f);
    dst[i] = v;
  }
  __syncthreads();
}

template <int NP>
__device__ inline void set_ident(float* A) {
  for (int i = threadIdx.x; i < NP * NP; i += blockDim.x)
    A[i] = ((i / NP) == (i % NP)) ? 1.f : 0.f;
  __syncthreads();
}

__device__ inline void ew_scale(float* A, int n, float s) {
  for (int i = threadIdx.x; i < n; i += blockDim.x) A[i] *= s;
  __syncthreads();
}

// ---------------------------------------------------------------------------
// Coupled Newton-Schulz (inverse-free Denman-Beavers):
//   y holds A (SPD, identity-padded) on entry; on exit y=sqrt(A), z=invsqrt(A).
//   T = 1.5I - 0.5 Z Y;  Y <- Y T;  Z <- T Z.   3 WMMA GEMMs / iter.
// ---------------------------------------------------------------------------
template <int NP, int NW>
__device__ inline void ns_sqrt(float*& y, float*& z, float*& t, float*& p,
                               float* red, int iters) {
  constexpr int NN = NP * NP;
  constexpr int TT = NP / 16;
  float c = trace_n<NP, NP>(y, red);
  c = fmaxf(c, 1.0f);
  float ic = 1.f / c;
  for (int i = threadIdx.x; i < NN; i += blockDim.x) {
    int r = i / NP, cc = i % NP;
    y[i] *= ic;
    z[i] = (r == cc) ? 1.f : 0.f;
  }
  __syncthreads();
  for (int it = 0; it < iters; ++it) {
    mmt<TT, TT, NP, NW, false, false>(z, NP, y, NP, t, NP, -0.5f, 1.5f);
    mmt<TT, TT, NP, NW, false, false>(y, NP, t, NP, p, NP, 1.f, 0.f);
    { float* tmp = y; y = p; p = tmp; }
    mmt<TT, TT, NP, NW, false, false>(t, NP, z, NP, p, NP, 1.f, 0.f);
    { float* tmp = z; z = p; p = tmp; }
  }
  float sc = sqrtf(c), isc = 1.f / sc;
  for (int i = threadIdx.x; i < NN; i += blockDim.x) {
    y[i] *= sc;
    z[i] *= isc;
  }
  __syncthreads();
}

// ---------------------------------------------------------------------------
// SPD matrix-function kernel: one workgroup per matrix, all buffers in LDS.
// modes: M_SQRT (out0=sqrt, out1=invsqrt), M_LOG, M_EXP, M_REEIG.
// Output index remap supports writing into a channel-concatenated tensor.
// ---------------------------------------------------------------------------
template <int N, int NP, int NW>
__global__ __launch_bounds__(NW * 32) void spd_func_kernel(
    const float* __restrict__ in, float* __restrict__ out0,
    float* __restrict__ out1, int mode, int in_ch, int out_ch_total,
    int out_ch_base) {
  extern __shared__ float sm[];
  constexpr int NN = NP * NP;
  constexpr int TT = NP / 16;
  float* b0 = sm;
  float* b1 = sm + NN;
  float* b2 = sm + 2 * NN;
  float* b3 = sm + 3 * NN;
  float* red = sm + 4 * NN;

  const int g = blockIdx.x;
  const int gi = (g / in_ch) * out_ch_total + out_ch_base + (g % in_ch);
  const float* src = in + (size_t)g * N * N;
  float* dst0 = out0 + (size_t)gi * N * N;

  float* y = b0;
  float* z = b1;
  float* t = b2;
  float* p = b3;

  if (mode == M_SQRT) {
    load_pad<N, NP>(y, src, 1.f);
    ns_sqrt<NP, NW>(y, z, t, p, red, 18);
    store_sym<N, NP>(dst0, y, 1.f, 0.f);
    if (out1) store_sym<N, NP>(out1 + (size_t)gi * N * N, z, 1.f, 0.f);
  } else if (mode == M_LOG) {
    // log A = 2^s log(A^(1/2^s)) + ln(c) I ; roots via Newton-Schulz.
    load_pad<N, NP>(y, src, 1.f);
    float c = trace_n<NP, NP>(y, red);
    c = fmaxf(c, 1.f);
    ew_scale(y, NN, 1.f / c);
    const int S = 4;
    for (int s = 0; s < S; ++s) ns_sqrt<NP, NW>(y, z, t, p, red, 14);
    float* m = z;  // M = Y - I (small)
    for (int i = threadIdx.x; i < NN; i += blockDim.x) {
      int r = i / NP, cc = i % NP;
      m[i] = y[i] - ((r == cc) ? 1.f : 0.f);
    }
    __syncthreads();
    float* h = p;  // Horner for log(1+M), 7 terms
    for (int i = threadIdx.x; i < NN; i += blockDim.x) {
      int r = i / NP, cc = i % NP;
      h[i] = (r == cc) ? (1.f / 7.f) : 0.f;
    }
    __syncthreads();
    for (int k = 6; k >= 1; --k) {
      float ck = ((k & 1) ? 1.f : -1.f) / (float)k;
      mmt<TT, TT, NP, NW, false, false>(m, NP, h, NP, t, NP, 1.f, ck);
      { float* tmp = h; h = t; t = tmp; }
    }
    mmt<TT, TT, NP, NW, false, false>(m, NP, h, NP, t, NP, 1.f, 0.f);
    store_sym<N, NP>(dst0, t, (float)(1 << S), logf(c));
  } else if (mode == M_EXP) {
    // scaling-and-squaring + 8-term Taylor Horner.
    load_pad<N, NP>(y, src, 0.f);
    const int S = 5;
    ew_scale(y, NN, 1.f / (float)(1 << S));
    float* h = p;
    set_ident<NP>(h);
    for (int k = 8; k >= 1; --k) {
      mmt<TT, TT, NP, NW, false, false>(y, NP, h, NP, t, NP, 1.f / (float)k, 1.f);
      { float* tmp = h; h = t; t = tmp; }
    }
    for (int s = 0; s < S; ++s) {
      mmt<TT, TT, NP, NW, false, false>(h, NP, h, NP, t, NP, 1.f, 0.f);
      { float* tmp = h; h = t; t = tmp; }
    }
    store_sym<N, NP>(dst0, h, 1.f, 0.f);
  } else {  // M_REEIG: clip(w,eps) = 0.5(A + eps I + |A - eps I|), |M|=sign(M)M
    load_pad<N, NP>(y, src, 1.f);
    for (int i = threadIdx.x; i < NN; i += blockDim.x) {
      int r = i / NP, cc = i % NP;
      if (r == cc) y[i] -= REEIG_EPS;  // y = M
    }
    __syncthreads();
    float part = 0.f;
    for (int i = threadIdx.x; i < NN; i += blockDim.x) part += y[i] * y[i];
    float nu = sqrtf(block_sum(part, red)) + 1e-20f;
    float* sgn = z;
    for (int i = threadIdx.x; i < NN; i += blockDim.x) sgn[i] = y[i] / nu;
    __syncthreads();
    for (int it = 0; it < 16; ++it) {  // Newton-Schulz sign iteration
      mmt<TT, TT, NP, NW, false, false>(sgn, NP, sgn, NP, t, NP, -0.5f, 1.5f);
      mmt<TT, TT, NP, NW, false, false>(sgn, NP, t, NP, p, NP, 1.f, 0.f);
      { float* tmp = sgn; sgn = p; p = tmp; }
    }
    mmt<TT, TT, NP, NW, false, false>(sgn, NP, y, NP, t, NP, 1.f, 0.f);
    for (int i = threadIdx.x; i < NN; i += blockDim.x)
      t[i] = 0.5f * (y[i] + t[i]);
    __syncthreads();
    store_sym<N, NP>(dst0, t, 1.f, REEIG_EPS);
  }
}

// ---------------------------------------------------------------------------
// bimap / congruence: Y[b,c] = sym( W X[b, c%ci] W^T ), W per-channel or
// per-(batch,channel) (used for bary2's  As M As / Ais B Ais products).
// ---------------------------------------------------------------------------
template <int NIN, int NIP, int NOUT, int NOP, int NW>
__global__ __launch_bounds__(NW * 32) void bimap_kernel(
    const float* __restrict__ X, const float* __restrict__ Wm,
    float* __restrict__ Y, int ci, int co, int w_per_batch) {
  extern __shared__ float sm[];
  float* Xs = sm;                   // NIP*NIP
  float* Ws = Xs + NIP * NIP;       // NOP*NIP
  float* T1 = Ws + NOP * NIP;       // NOP*NIP
  float* Ys = T1 + NOP * NIP;       // NOP*NOP
  const int g = blockIdx.x;
  const int b = g / co, c = g % co;
  const float* src = X + ((size_t)b * ci + (c % ci)) * NIN * NIN;
  const float* wsrc =
      Wm + (size_t)(w_per_batch ? (b * co + c) : c) * NOUT * NIN;
  for (int i = threadIdx.x; i < NIP * NIP; i += blockDim.x) {
    int r = i / NIP, cc = i % NIP;
    Xs[i] = (r < NIN && cc < NIN) ? src[r * NIN + cc] : 0.f;
  }
  for (int i = threadIdx.x; i < NOP * NIP; i += blockDim.x) {
    int r = i / NIP, cc = i % NIP;
    Ws[i] = (r < NOUT && cc < NIN) ? wsrc[r * NIN + cc] : 0.f;
  }
  __syncthreads();
  mmt<NOP / 16, NIP / 16, NIP, NW, false, false>(Ws, NIP, Xs, NIP, T1, NIP,
                                                 1.f, 0.f);
  mmt<NOP / 16, NOP / 16, NIP, NW, false, true>(T1, NIP, Ws, NIP, Ys, NOP, 1.f,
                                                0.f);
  float* dst = Y + ((size_t)b * co + c) * NOUT * NOUT;
  for (int i = threadIdx.x; i < NOUT * NOUT; i += blockDim.x) {
    int r = i / NOUT, cc = i % NOUT;
    dst[i] = 0.5f * (Ys[r * NOP + cc] + Ys[cc * NOP + r]);
  }
}

// ---------------------------------------------------------------------------
// Front-end: f = x^T fc_w^T + b (100x100 per batch, K=512 on WMMA),
// center over features, cov = f^T f/(T-1) (WMMA), + ridge.
// Next K-panel of x / fc_w is prefetched (global_prefetch_b8) while the
// current panel computes; panels are L2-resident.
// ---------------------------------------------------------------------------
__global__ __launch_bounds__(256) void cov_kernel(const float* __restrict__ x,
                                                  const float* __restrict__ fc_w,
                                                  const float* __restrict__ fc_b,
                                                  float* __restrict__ cov) {
  constexpr int NP = 112, T = 100, N = 100, K = 512, KC = 16;
  extern __shared__ float sm[];
  float* F = sm;                 // NP*NP  (f: T x N, padded)
  float* C = F + NP * NP;        // NP*NP
  float* As = C + NP * NP;       // NP*KC
  float* Bs = As + NP * KC;      // KC*NP
  float* red = Bs + KC * NP;
  const int b = blockIdx.x;
  const int lane = threadIdx.x & 31, wave = threadIdx.x >> 5;
  const int half = lane >> 4, li = lane & 15;
  const float* xb = x + (size_t)b * K * T;  // x[b][k][t]

  v8f acc[7];
#pragma unroll
  for (int i = 0; i < 7; ++i) acc[i] = {0.f, 0.f, 0.f, 0.f, 0.f, 0.f, 0.f, 0.f};

  for (int k0 = 0; k0 < K; k0 += KC) {
    __syncthreads();
    for (int i = threadIdx.x; i < NP * KC; i += 256) {
      int t = i / KC, kk = i % KC;
      As[i] = (t < T) ? xb[(size_t)(k0 + kk) * T + t] : 0.f;
    }
    for (int i = threadIdx.x; i < KC * NP; i += 256) {
      int kk = i / NP, n = i % NP;
      Bs[i] = (n < N) ? fc_w[(size_t)n * K + k0 + kk] : 0.f;
    }
    if (k0 + KC < K) {  // prefetch next panel
      __builtin_prefetch(&xb[(size_t)(k0 + KC) * T + (threadIdx.x & 63) * 16], 0, 0);
      __builtin_prefetch(&fc_w[(size_t)(threadIdx.x & 63) * K + k0 + KC], 0, 0);
    }
    __syncthreads();
#pragma unroll
    for (int ti = 0; ti < 7; ++ti) {
      int t = wave + ti * 8;  // wave-uniform -> EXEC all-1 around WMMA
      if (t < 49) {
        int tr = t / 7, tc = t % 7;
        v2f a[4], bb[4];
#pragma unroll
        for (int u = 0; u < 4; ++u) {
          int kk = 4 * u;
          int ka = kk + 2 * half;
          a[u].x = As[(tr * 16 + li) * KC + ka];
          a[u].y = As[(tr * 16 + li) * KC + ka + 1];
          bb[u].x = Bs[(kk + half) * NP + tc * 16 + li];
          bb[u].y = Bs[(kk + 2 + half) * NP + tc * 16 + li];
        }
#pragma unroll
        for (int u = 0; u < 4; ++u) acc[ti] = wmma4(a[u], bb[u], acc[ti]);
      }
    }
  }
  __syncthreads();
#pragma unroll
  for (int ti = 0; ti < 7; ++ti) {
    int t = wave + ti * 8;
    if (t < 49) {
      int tr = t / 7, tc = t % 7;
#pragma unroll
      for (int r = 0; r < 8; ++r)
        F[(tr * 16 + r + 8 * half) * NP + tc * 16 + li] = acc[ti][r];
    }
  }
  __syncthreads();
  // bias + center over feature axis (n), zero the pad
  for (int t = threadIdx.x; t < NP; t += 256) {
    if (t < T) {
      float mu = 0.f;
      for (int n = 0; n < N; ++n) {
        float v = F[t * NP + n] + fc_b[n];
        F[t * NP + n] = v;
        mu += v;
      }
      mu /= (float)N;
      for (int n = 0; n < N; ++n) F[t * NP + n] -= mu;
      for (int n = N; n < NP; ++n) F[t * NP + n] = 0.f;
    } else {
      for (int n = 0; n < NP; ++n) F[t * NP + n] = 0.f;
    }
  }
  __syncthreads();
  mmt<7, 7, NP, 8, true, false>(F, NP, F, NP, C, NP, 1.f / (float)(T - 1), 0.f);
  float part = 0.f;
  for (int i = threadIdx.x; i < N; i += 256) part += C[i * NP + i];
  float lam = 1e-5f * block_sum(part, red);
  float* dst = cov + (size_t)b * N * N;
  for (int i = threadIdx.x; i < N * N; i += 256) {
    int r = i / N, cc = i % N;
    dst[i] = C[r * NP + cc] + ((r == cc) ? lam : 0.f);
  }
}

// mean over batch of logm outputs (per channel)
__global__ void mean_kernel(const float* __restrict__ in,
                            float* __restrict__ out, int B, int C, int NN) {
  int idx = blockIdx.x * blockDim.x + threadIdx.x;
  if (idx >= C * NN) return;
  int c = idx / NN, i = idx % NN;
  float s = 0.f;
  for (int b = 0; b < B; ++b) s += in[((size_t)b * C + c) * NN + i];
  out[idx] = s / (float)B;
}

// classifier: out(64x100) = L(64x20000) @ cls_w^T + b, WMMA tiled, K staged in
// LDS, next panel prefetched.
__global__ __launch_bounds__(32) void cls_kernel(const float* __restrict__ L,
                                                 const float* __restrict__ cls_w,
                                                 const float* __restrict__ cls_b,
                                                 float* __restrict__ out) {
  constexpr int N = 100, K = 20000, KC = 32;
  __shared__ float As[16 * KC];
  __shared__ float Bs[KC * 16];
  const int mt = blockIdx.x / 7;
  const int nt = blockIdx.x % 7;
  const int lane = threadIdx.x;
  const int half = lane >> 4, li = lane & 15;
  v8f acc = {0.f, 0.f, 0.f, 0.f, 0.f, 0.f, 0.f, 0.f};
  for (int k0 = 0; k0 < K; k0 += KC) {
    __syncthreads();
    for (int i = lane; i < 16 * KC; i += 32) {
      int m = i / KC, kk = i % KC;
      As[i] = L[(size_t)(mt * 16 + m) * K + k0 + kk];
    }
    for (int i = lane; i < KC * 16; i += 32) {
      int kk = i / 16, n = i % 16;
      int gn = nt * 16 + n;
      Bs[i] = (gn < N) ? cls_w[(size_t)gn * K + k0 + kk] : 0.f;
    }
    if (k0 + KC < K) {  // prefetch next panel
      __builtin_prefetch(&L[(size_t)(mt * 16 + (lane & 15)) * K + k0 + KC], 0, 0);
      int gn = nt * 16 + (lane & 15);
      if (gn < N) __builtin_prefetch(&cls_w[(size_t)gn * K + k0 + KC], 0, 0);
    }
    __syncthreads();
    v2f a[8], b[8];
#pragma unroll
    for (int u = 0; u < 8; ++u) {
      int kk = 4 * u;
      int ka = kk + 2 * half;
      a[u].x = As[li * KC + ka];
      a[u].y = As[li * KC + ka + 1];
      b[u].x = Bs[(kk + half) * 16 + li];
      b[u].y = Bs[(kk + 2 + half) * 16 + li];
    }
#pragma unroll
    for (int u = 0; u < 8; ++u) acc = wmma4(a[u], b[u], acc);
  }
#pragma unroll
  for (int r = 0; r < 8; ++r) {
    int row = mt * 16 + r + 8 * half;
    int col = nt * 16 + li;
    if (col < N) out[(size_t)row * N + col] = acc[r] + cls_b[col];
  }
}

// ---------------------------------------------------------------------------
extern "C" void kernel_launch(void* const* d_in, const int* in_sizes, int n_in,
                              void* d_out, int out_size, void* d_ws,
                              size_t ws_size, hipStream_t stream) {
  (void)in_sizes; (void)n_in; (void)out_size; (void)ws_size;
  const float* x = (const float*)d_in[0];
  const float* fc_w = (const float*)d_in[1];
  const float* fc_b = (const float*)d_in[2];
  const float* stem_w = (const float*)d_in[3];
  const float* pre0_w = (const float*)d_in[4];
  const float* pre1_w = (const float*)d_in[5];
  const float* wr0 = (const float*)d_in[6];
  const float* wr1 = (const float*)d_in[7];
  const float* wn2 = (const float*)d_in[8];
  const float* wn4 = (const float*)d_in[9];
  const float* wn7 = (const float*)d_in[10];
  const float* cls_w = (const float*)d_in[11];
  const float* cls_b = (const float*)d_in[12];
  float* out = (float*)d_out;
  float* W = (float*)d_ws;

  // workspace layout (floats)
  float* COVp = W + 0;              // 64*10000
  float* Sp = W + 640000;           // 64*10000
  float* LBUF = W + 1280000;        // up to 64*2*10000
  float* GBUF = W + 2560000;        // 2*10000
  float* GEXP = W + 2580000;
  float* GISQ = W + 2600000;
  float* GJUNK = W + 2620000;
  float* RBUF = W + 2640000;        // 64*10000
  float* S0p = W + 3280000;         // 64*2*10000
  float* S1p = W + 4560000;
  float* A2p = W + 5840000;         // 64*2*2500 each from here on
  float* B2p = W + 6160000;
  float* TAs = W + 6480000;
  float* TAis = W + 6800000;
  float* TM = W + 7120000;
  float* TMs = W + 7440000;
  float* TJ = W + 7760000;
  float* ST2p = W + 8080000;
  float* ST3p = W + 8400000;
  float* ST4p = W + 8720000;
  float* ST5p = W + 9040000;
  float* LCAT = W + 9360000;        // 64*8*2500

  const size_t SH_SPD100 = (4 * 112 * 112 + 32) * sizeof(float);
  const size_t SH_SPD50 = (4 * 64 * 64 + 32) * sizeof(float);
  const size_t SH_BM100 = (size_t)(4 * 112 * 112) * sizeof(float);
  const size_t SH_BM100_50 =
      (size_t)(112 * 112 + 2 * 64 * 112 + 64 * 64) * sizeof(float);
  const size_t SH_BM50 = (size_t)(4 * 64 * 64) * sizeof(float);
  const size_t SH_COV = (size_t)(2 * 112 * 112 + 2 * 112 * 16 + 32) * sizeof(float);

  auto bn100 = [&](float* Xio, int C) {  // batchnorm_spd, 100x100, C channels
    spd_func_kernel<100, 112, 8><<<64 * C, 256, SH_SPD100, stream>>>(
        Xio, LBUF, nullptr, M_LOG, C, C, 0);
    int total = C * 10000;
    mean_kernel<<<(total + 255) / 256, 256, 0, stream>>>(LBUF, GBUF, 64, C, 10000);
    spd_func_kernel<100, 112, 8><<<C, 256, SH_SPD100, stream>>>(
        GBUF, GEXP, nullptr, M_EXP, 1, 1, 0);
    spd_func_kernel<100, 112, 8><<<C, 256, SH_SPD100, stream>>>(
        GEXP, GJUNK, GISQ, M_SQRT, 1, 1, 0);
    bimap_kernel<100, 112, 100, 112, 8><<<64 * C, 256, SH_BM100, stream>>>(
        Xio, GISQ, Xio, C, C, 0);
  };
  auto bary2_50 = [&](const float* A, const float* Bm, float* OUT) {
    spd_func_kernel<50, 64, 4><<<128, 128, SH_SPD50, stream>>>(
        A, TAs, TAis, M_SQRT, 2, 2, 0);
    bimap_kernel<50, 64, 50, 64, 4><<<128, 128, SH_BM50, stream>>>(
        Bm, TAis, TM, 2, 2, 1);  // M = sym(Ais B Ais)
    spd_func_kernel<50, 64, 4><<<128, 128, SH_SPD50, stream>>>(
        TM, TMs, TJ, M_SQRT, 2, 2, 0);
    bimap_kernel<50, 64, 50, 64, 4><<<128, 128, SH_BM50, stream>>>(
        TMs, TAs, OUT, 2, 2, 1);  // sym(As Ms As)
  };

  // 1) covariance front-end
  cov_kernel<<<64, 256, SH_COV, stream>>>(x, fc_w, fc_b, COVp);
  // 2) stem bimap + batchnorm
  bimap_kernel<100, 112, 100, 112, 8><<<64, 256, SH_BM100, stream>>>(
      COVp, stem_w, Sp, 1, 1, 0);
  bn100(Sp, 1);
  // 3) reeig(s) once, two pre-branches
  spd_func_kernel<100, 112, 8><<<64, 256, SH_SPD100, stream>>>(
      Sp, RBUF, nullptr, M_REEIG, 1, 1, 0);
  bimap_kernel<100, 112, 100, 112, 8><<<128, 256, SH_BM100, stream>>>(
      RBUF, pre0_w, S0p, 1, 2, 0);
  bn100(S0p, 2);
  bimap_kernel<100, 112, 100, 112, 8><<<128, 256, SH_BM100, stream>>>(
      RBUF, pre1_w, S1p, 1, 2, 0);
  bn100(S1p, 2);
  // 4) reduce to 50x50 and barycenter chain
  bimap_kernel<100, 112, 50, 64, 8><<<128, 256, SH_BM100_50, stream>>>(
      S0p, wr0, A2p, 2, 2, 0);
  bimap_kernel<100, 112, 50, 64, 8><<<128, 256, SH_BM100_50, stream>>>(
      S1p, wr1, B2p, 2, 2, 0);
  bary2_50(A2p, B2p, ST2p);
  bimap_kernel<50, 64, 50, 64, 4><<<128, 128, SH_BM50, stream>>>(
      ST2p, wn2, A2p, 2, 2, 0);
  bary2_50(A2p, ST2p, ST3p);
  bimap_kernel<50, 64, 50, 64, 4><<<128, 128, SH_BM50, stream>>>(
      ST3p, wn4, A2p, 2, 2, 0);
  bary2_50(A2p, ST2p, ST4p);
  bimap_kernel<50, 64, 50, 64, 4><<<128, 128, SH_BM50, stream>>>(
      ST3p, wn7, B2p, 2, 2, 0);
  bary2_50(ST4p, B2p, ST5p);
  // 5) logm into channel-concatenated buffer (B,8,50,50)
  spd_func_kernel<50, 64, 4><<<128, 128, SH_SPD50, stream>>>(
      ST2p, LCAT, nullptr, M_LOG, 2, 8, 0);
  spd_func_kernel<50, 64, 4><<<128, 128, SH_SPD50, stream>>>(
      ST3p, LCAT, nullptr, M_LOG, 2, 8, 2);
  spd_func_kernel<50, 64, 4><<<128, 128, SH_SPD50, stream>>>(
      ST4p, LCAT, nullptr, M_LOG, 2, 8, 4);
  spd_func_kernel<50, 64, 4><<<128, 128, SH_SPD50, stream>>>(
      ST5p, LCAT, nullptr, M_LOG, 2, 8, 6);
  // 6) classifier
  cls_kernel<<<28, 32, 0, stream>>>(LCAT, cls_w, cls_b, out);
}